// Arma_84086869721212
// MI455X (gfx1250) — compile-verified
//
#include <hip/hip_runtime.h>
#include <math.h>

typedef float v2f __attribute__((ext_vector_type(2)));
typedef float v8f __attribute__((ext_vector_type(8)));

#define DIMF 32
#define KSTACK 3

// ---------------------------------------------------------------- utilities
__global__ void zero_kernel(float* __restrict__ p, int n) {
    int i = blockIdx.x * blockDim.x + threadIdx.x;
    if (i < n) p[i] = 0.0f;
}

__global__ void deg_kernel(const int* __restrict__ dst, float* __restrict__ deg, int E) {
    int e = blockIdx.x * blockDim.x + threadIdx.x;
    if (e < E) atomicAdd(&deg[dst[e]], 1.0f);
}

__global__ void norm_kernel(const int* __restrict__ src, const int* __restrict__ dst,
                            const float* __restrict__ deg, float* __restrict__ nrm, int E) {
    int e = blockIdx.x * blockDim.x + threadIdx.x;
    if (e >= E) return;
    float ds = deg[src[e]];
    float dd = deg[dst[e]];
    float is = ds > 0.0f ? rsqrtf(ds) : 0.0f;
    float id = dd > 0.0f ? rsqrtf(dd) : 0.0f;
    nrm[e] = is * id;
}

// ---------------------------------------------------------------- WMMA GEMM
// C[b] = X_b @ W[b];  X_b = X + b*xBatchStride (0 => shared X)
// X row-major [*, Fin], W [B, Fin, Fout], C [B, N, Fout].
// One wave computes one 16x16 output tile with V_WMMA_F32_16X16X4_F32.
__global__ void gemm_wmma(const float* __restrict__ X, int xBatchStride,
                          const float* __restrict__ W, float* __restrict__ C,
                          int N, int Fin, int Fout,
                          int tilesPerBatch, int colTiles, int totalTiles) {
    int wave = (blockIdx.x * blockDim.x + threadIdx.x) >> 5;
    if (wave >= totalTiles) return;               // wave-uniform guard (EXEC stays all-1)
    int lane = threadIdx.x & 31;

    int b   = wave / tilesPerBatch;
    int rem = wave - b * tilesPerBatch;
    int rt  = rem / colTiles;
    int ct  = rem - rt * colTiles;

    int half  = lane >> 4;        // 0: K=0,1   1: K=2,3
    int l15   = lane & 15;
    int row   = rt * 16 + l15;    // A row (M)
    int col   = ct * 16 + l15;    // B/C col (N)
    int rowc  = row < N ? row : N - 1;
    int khalf = half * 2;

    const float* Xb   = X + (long)b * xBatchStride;
    const float* Wb   = W + (long)b * Fin * Fout;
    const float* xrow = Xb + (long)rowc * Fin + khalf;   // even offset -> 8B aligned
    const float* wcol = Wb + (long)khalf * Fout + col;

    v8f acc = {};
    for (int k = 0; k < Fin; k += 4) {
        v2f a = *(const v2f*)(xrow + k);   // A: {X[row,k+khalf], X[row,k+khalf+1]}
        v2f bm;
        bm.x = wcol[k * Fout];             // B: {W[k+khalf,col], W[k+khalf+1,col]}
        bm.y = wcol[(k + 1) * Fout];
        acc = __builtin_amdgcn_wmma_f32_16x16x4_f32(
            /*neg_a=*/false, a, /*neg_b=*/false, bm,
            /*c_mod=*/(short)0, acc, /*reuse_a=*/false, /*reuse_b=*/false);
    }

    // C/D layout: VGPR r -> M = r (lanes 0-15) / r+8 (lanes 16-31), N = lane&15
    float* Cb = C + (long)b * N * Fout;
    int m0 = rt * 16 + half * 8;
#pragma unroll
    for (int r = 0; r < 8; ++r) {
        int m = m0 + r;
        if (m < N) Cb[(long)m * Fout + col] = acc[r];
    }
}

// ---------------------------------------------------------------- scatter
// acc[k,dst,f] += src_feat[k,src,f] * norm[e]; one wave per edge, lane = feature.
__global__ void prop_kernel(const float* __restrict__ src_feat, float* __restrict__ acc,
                            const int* __restrict__ srcIdx, const int* __restrict__ dstIdx,
                            const float* __restrict__ nrm, int E, int NF) {
    int gid = blockIdx.x * blockDim.x + threadIdx.x;
    int e = gid >> 5;
    if (e >= E) return;
    int f = gid & 31;
    int s = srcIdx[e] * DIMF + f;
    int d = dstIdx[e] * DIMF + f;
    float w = nrm[e];
#pragma unroll
    for (int k = 0; k < KSTACK; ++k)
        atomicAdd(&acc[k * NF + d], src_feat[k * NF + s] * w);
}

// ---------------------------------------------------------------- epilogues
__global__ void combine_kernel(const float* __restrict__ P, const float* __restrict__ R,
                               const float* __restrict__ bias, float* __restrict__ out,
                               int total, int NF, int doRelu) {
    int i = blockIdx.x * blockDim.x + threadIdx.x;
    if (i >= total) return;
    int k = i / NF;
    int f = i & (DIMF - 1);
    float v = P[i] + R[i] + bias[k * DIMF + f];
    if (doRelu) v = fmaxf(v, 0.0f);
    out[i] = v;
}

__global__ void mean_kernel(const float* __restrict__ in, float* __restrict__ out,
                            int NF, int doRelu) {
    int i = blockIdx.x * blockDim.x + threadIdx.x;
    if (i >= NF) return;
    float v = (in[i] + in[i + NF] + in[i + 2 * NF]) * (1.0f / 3.0f);
    if (doRelu) v = fmaxf(v, 0.0f);
    out[i] = v;
}

// ---------------------------------------------------------------- driver
extern "C" void kernel_launch(void* const* d_in, const int* in_sizes, int n_in,
                              void* d_out, int out_size, void* d_ws, size_t ws_size,
                              hipStream_t stream) {
    (void)n_in; (void)out_size; (void)ws_size;
    const float* x       = (const float*)d_in[0];
    const int*   edge    = (const int*)  d_in[1];
    const float* w1_init = (const float*)d_in[2];
    const float* w1      = (const float*)d_in[3];
    const float* w1_root = (const float*)d_in[4];
    const float* b1      = (const float*)d_in[5];
    const float* w2_init = (const float*)d_in[6];
    const float* w2      = (const float*)d_in[7];
    const float* w2_root = (const float*)d_in[8];
    const float* b2      = (const float*)d_in[9];

    const int NUM_FEAT = 64;
    int N = in_sizes[0] / NUM_FEAT;     // 100000
    int E = in_sizes[1] / 2;            // 1600000
    const int* srcIdx = edge;
    const int* dstIdx = edge + E;

    int NF  = N * DIMF;                 // per-stack feature count
    int KNF = KSTACK * NF;

    float* ws   = (float*)d_ws;
    float* deg  = ws;
    float* nrm  = deg  + N;
    float* bufA = nrm  + E;
    float* bufB = bufA + KNF;
    float* bufR = bufB + KNF;
    float* h    = bufR + KNF;

    const int TB = 256;
    auto nb = [](long n, int tb) { return (int)((n + tb - 1) / tb); };

    // gcn_norm (no self loops)
    zero_kernel<<<nb(N, TB), TB, 0, stream>>>(deg, N);
    deg_kernel <<<nb(E, TB), TB, 0, stream>>>(dstIdx, deg, E);
    norm_kernel<<<nb(E, TB), TB, 0, stream>>>(srcIdx, dstIdx, deg, nrm, E);

    int rowTiles      = (N + 15) / 16;
    int colTiles      = DIMF / 16;                  // 2
    int tilesPerBatch = rowTiles * colTiles;
    int totalTiles    = tilesPerBatch * KSTACK;
    int gemmBlocks    = nb((long)totalTiles * 32, TB);
    int propBlocks    = nb((long)E * 32, TB);

    // ---------------- layer 1 (act = relu) ----------------
    gemm_wmma<<<gemmBlocks, TB, 0, stream>>>(x, 0, w1_init, bufA, N, NUM_FEAT, DIMF,
                                             tilesPerBatch, colTiles, totalTiles);
    gemm_wmma<<<gemmBlocks, TB, 0, stream>>>(x, 0, w1_root, bufR, N, NUM_FEAT, DIMF,
                                             tilesPerBatch, colTiles, totalTiles);
    // t = 0
    zero_kernel<<<nb(KNF, TB), TB, 0, stream>>>(bufB, KNF);
    prop_kernel<<<propBlocks, TB, 0, stream>>>(bufA, bufB, srcIdx, dstIdx, nrm, E, NF);
    combine_kernel<<<nb(KNF, TB), TB, 0, stream>>>(bufB, bufR, b1, bufA, KNF, NF, 1);
    // t = 1:  out = out @ w1[0]  (batched X), then prop + relu
    gemm_wmma<<<gemmBlocks, TB, 0, stream>>>(bufA, NF, w1, bufB, N, DIMF, DIMF,
                                             tilesPerBatch, colTiles, totalTiles);
    zero_kernel<<<nb(KNF, TB), TB, 0, stream>>>(bufA, KNF);
    prop_kernel<<<propBlocks, TB, 0, stream>>>(bufB, bufA, srcIdx, dstIdx, nrm, E, NF);
    combine_kernel<<<nb(KNF, TB), TB, 0, stream>>>(bufA, bufR, b1, bufB, KNF, NF, 1);
    // h = relu(mean over stacks)
    mean_kernel<<<nb(NF, TB), TB, 0, stream>>>(bufB, h, NF, 1);

    // ---------------- layer 2 (act = identity) ----------------
    gemm_wmma<<<gemmBlocks, TB, 0, stream>>>(h, 0, w2_init, bufA, N, DIMF, DIMF,
                                             tilesPerBatch, colTiles, totalTiles);
    gemm_wmma<<<gemmBlocks, TB, 0, stream>>>(h, 0, w2_root, bufR, N, DIMF, DIMF,
                                             tilesPerBatch, colTiles, totalTiles);
    // t = 0
    zero_kernel<<<nb(KNF, TB), TB, 0, stream>>>(bufB, KNF);
    prop_kernel<<<propBlocks, TB, 0, stream>>>(bufA, bufB, srcIdx, dstIdx, nrm, E, NF);
    combine_kernel<<<nb(KNF, TB), TB, 0, stream>>>(bufB, bufR, b2, bufA, KNF, NF, 0);
    // t = 1
    gemm_wmma<<<gemmBlocks, TB, 0, stream>>>(bufA, NF, w2, bufB, N, DIMF, DIMF,
                                             tilesPerBatch, colTiles, totalTiles);
    zero_kernel<<<nb(KNF, TB), TB, 0, stream>>>(bufA, KNF);
    prop_kernel<<<propBlocks, TB, 0, stream>>>(bufB, bufA, srcIdx, dstIdx, nrm, E, NF);
    combine_kernel<<<nb(KNF, TB), TB, 0, stream>>>(bufA, bufR, b2, bufB, KNF, NF, 0);
    // out = mean over stacks
    mean_kernel<<<nb(NF, TB), TB, 0, stream>>>(bufB, (float*)d_out, NF, 0);
}